// ModelNew_14783277433190
// MI455X (gfx1250) — compile-verified
//
#include <hip/hip_runtime.h>

// Reverse cumsum along rows: 2048 rows x 32768 f32.
// Memory-bound (537 MB traffic -> ~23us floor @ 23.3 TB/s).
// One 1024-thread block (32 wave32 waves) per row; 8 tiles of 4096 floats,
// processed right-to-left with a serial carry; async global->LDS double buffer.

typedef __attribute__((ext_vector_type(4))) float f32x4;
typedef __attribute__((ext_vector_type(4))) int v4i;

#ifndef __has_builtin
#define __has_builtin(x) 0
#endif

#if __has_builtin(__builtin_amdgcn_global_load_async_to_lds_b128)
#define USE_ASYNC_LDS 1
#else
#define USE_ASYNC_LDS 0
#endif

#if __has_builtin(__builtin_amdgcn_s_wait_asynccnt)
#define WAIT_ASYNCCNT(n) __builtin_amdgcn_s_wait_asynccnt(n)
#else
#define WAIT_ASYNCCNT(n) asm volatile("s_wait_asynccnt %0" ::"n"(n) : "memory")
#endif

#if USE_ASYNC_LDS
typedef __attribute__((address_space(1))) v4i* as1_v4i_ptr;
typedef __attribute__((address_space(3))) v4i* as3_v4i_ptr;

__device__ __forceinline__ void async_copy16(const float* g, float* l) {
  // Each lane copies 16B global -> LDS via the async DMA path (ASYNCcnt).
  __builtin_amdgcn_global_load_async_to_lds_b128(
      (as1_v4i_ptr)g, (as3_v4i_ptr)l, /*offset=*/0, /*cpol=*/0);
}
#endif

#define NCOL 32768
#define TILE 4096
#define NTHREADS 1024

__global__ __launch_bounds__(NTHREADS) void revcumsum_rows_kernel(
    const float* __restrict__ x, float* __restrict__ out) {
  const int t = threadIdx.x;          // 0..1023
  const int lane = t & 31;            // wave32 lane
  const int wave = t >> 5;            // 0..31
  const long row = blockIdx.x;
  const float* rowIn = x + row * (long)NCOL;
  float* rowOut = out + row * (long)NCOL;

#if USE_ASYNC_LDS
  __shared__ float tile[2][TILE];     // 32 KB double buffer
#endif
  __shared__ float sWaveTot[32];
  __shared__ float sWaveOff[32];
  __shared__ float sTileTot;

#if USE_ASYNC_LDS
  // Prime the pipeline: rightmost tile (index 7) into buffer 0.
  async_copy16(rowIn + 7 * TILE + t * 4, &tile[0][t * 4]);
#endif

  float carry = 0.0f;  // sum of all tiles to the right of the current one
  int buf = 0;

  for (int tl = 7; tl >= 0; --tl) {
    f32x4 v;
#if USE_ASYNC_LDS
    if (tl > 0) {
      // Prefetch the next (left-neighbor) tile into the other buffer.
      // Its previous readers are two __syncthreads() behind us -> no WAR.
      async_copy16(rowIn + (tl - 1) * TILE + t * 4, &tile[buf ^ 1][t * 4]);
      WAIT_ASYNCCNT(1);  // 2 outstanding; current tile is the older one
    } else {
      WAIT_ASYNCCNT(0);
    }
    asm volatile("" ::: "memory");
    v = *(const f32x4*)&tile[buf][t * 4];  // ds_load_b128, own 16B only
#else
    v = __builtin_nontemporal_load((const f32x4*)(rowIn + tl * TILE + t * 4));
#endif

    // In-thread reverse inclusive scan of 4 elements.
    const float s3 = v.w;
    const float s2 = v.z + s3;
    const float s1 = v.y + s2;
    const float s0 = v.x + s1;  // thread total

    // Wave32 suffix (reverse inclusive) scan of per-thread totals.
    float tot = s0;
#pragma unroll
    for (int off = 1; off < 32; off <<= 1) {
      float n = __shfl_down(tot, off, 32);
      if (lane + off < 32) tot += n;
    }
    const float laneExcl = tot - s0;  // sum of lanes strictly after this one

    if (lane == 0) sWaveTot[wave] = tot;  // wave total
    __syncthreads();

    // Cross-wave suffix scan of the 32 wave totals (done by wave 0).
    if (wave == 0) {
      const float wt = sWaveTot[lane];
      float wi = wt;
#pragma unroll
      for (int off = 1; off < 32; off <<= 1) {
        float n = __shfl_down(wi, off, 32);
        if (lane + off < 32) wi += n;
      }
      sWaveOff[lane] = wi - wt;       // sum of waves strictly after
      if (lane == 0) sTileTot = wi;   // whole-tile total
    }
    __syncthreads();

    const float base = laneExcl + sWaveOff[wave] + carry;
    f32x4 r;
    r.x = s0 + base;
    r.y = s1 + base;
    r.z = s2 + base;
    r.w = s3 + base;
    // Streaming store: written once, never re-read -> keep out of caches.
    __builtin_nontemporal_store(r, (f32x4*)(rowOut + tl * TILE + t * 4));

    carry += sTileTot;  // read before next iteration's first barrier
    buf ^= 1;
  }
}

extern "C" void kernel_launch(void* const* d_in, const int* in_sizes, int n_in,
                              void* d_out, int out_size, void* d_ws,
                              size_t ws_size, hipStream_t stream) {
  (void)n_in;
  (void)out_size;
  (void)d_ws;
  (void)ws_size;
  const float* x = (const float*)d_in[0];
  float* out = (float*)d_out;
  const int rows = (int)((long)in_sizes[0] / (long)NCOL);  // 2048
  dim3 grid(rows), block(NTHREADS);
  hipLaunchKernelGGL(revcumsum_rows_kernel, grid, block, 0, stream, x, out);
}